// OctreeDWConvNorm_9010841387711
// MI455X (gfx1250) — compile-verified
//
#include <hip/hip_runtime.h>

#define KNB 27
#define CH  64
#define EPSV 1e-5f

typedef float v2f __attribute__((ext_vector_type(2)));
typedef float v4f __attribute__((ext_vector_type(4)));
typedef float v8f __attribute__((ext_vector_type(8)));

static constexpr int NBLK1_MAX = 2048;   // partial-sum rows (power of 2, multiple of 4)

// ---------------------------------------------------------------------------
// Kernel 1: depthwise gather-conv + per-block BN partial sums (sum, sumsq).
// One wave per output row; lane l owns channels {2l, 2l+1} (float2, coalesced
// 256B per wave per neighbor). Gather data left cacheable (27x L2 reuse);
// neigh loads and out stores are non-temporal (streamed once).
// ---------------------------------------------------------------------------
__global__ __launch_bounds__(256) void dwconv_partial_kernel(
    const float* __restrict__ data, const float* __restrict__ w,
    const int* __restrict__ neigh, float* __restrict__ out,
    float* __restrict__ partials, int n, int nblk)
{
  __shared__ float lw[KNB * CH];     // weights tile: 6.75 KB
  __shared__ float red[2 * CH];      // block partials: 64 sums + 64 sumsq

  const int tid = threadIdx.x;
  for (int i = tid; i < KNB * CH; i += 256) lw[i] = w[i];
  if (tid < 2 * CH) red[tid] = 0.0f;
  __syncthreads();

  const int lane = tid & 31;
  const int wv   = tid >> 5;
  const int gw   = blockIdx.x * 8 + wv;  // global wave id
  const int nw   = nblk * 8;             // total waves

  float sx = 0.f, sy = 0.f, qx = 0.f, qy = 0.f;

  for (int n0 = gw; n0 < n; n0 += nw) {
    const long nb = (long)n0 * KNB;
    // lanes 0..26 fetch this row's neighbor indices (read-once -> NT)
    int myi = -1;
    if (lane < KNB) myi = __builtin_nontemporal_load(&neigh[nb + lane]);
    // speculative prefetch of next row's indices (global_prefetch_b8)
    if (lane < KNB && (n0 + nw) < n)
      __builtin_prefetch(&neigh[(long)(n0 + nw) * KNB + lane], 0, 1);

    float ax = 0.f, ay = 0.f;
#pragma unroll
    for (int k = 0; k < KNB; ++k) {
      const int idx = __shfl(myi, k, 32);        // wave32 broadcast
      if (idx >= 0) {
        const float2 g  = *(const float2*)&data[(long)idx * CH + 2 * lane];
        const float2 wk = *(const float2*)&lw[k * CH + 2 * lane];
        ax = fmaf(g.x, wk.x, ax);
        ay = fmaf(g.y, wk.y, ay);
      }
    }
    v2f o; o[0] = ax; o[1] = ay;
    __builtin_nontemporal_store(o, (v2f*)&out[(long)n0 * CH + 2 * lane]);

    sx += ax; sy += ay;
    qx += ax * ax; qy += ay * ay;
  }

  // cross-wave reduction inside the block (ds_add_f32 atomics)
  atomicAdd(&red[2 * lane],          sx);
  atomicAdd(&red[2 * lane + 1],      sy);
  atomicAdd(&red[CH + 2 * lane],     qx);
  atomicAdd(&red[CH + 2 * lane + 1], qy);
  __syncthreads();

  if (tid < 2 * CH) partials[blockIdx.x * (2 * CH) + tid] = red[tid];
}

// ---------------------------------------------------------------------------
// Kernel 2: WMMA column reduction of partials (nblk x 128) -> 128 column sums.
// A = ones(16x4) so D = column sums of accumulated B tiles; with all-ones A
// the result is invariant to the K-row<->lane permutation of the B layout.
// Wave w owns columns [16w, 16w+16); 8 waves cover 128 columns.
// Then compute scale = gamma*rsqrt(var+eps), bias = beta - mean*scale.
// ---------------------------------------------------------------------------
__global__ __launch_bounds__(256) void stats_wmma_kernel(
    const float* __restrict__ partials, const float* __restrict__ gamma,
    const float* __restrict__ beta, float* __restrict__ sb, int nblk, int n)
{
  __shared__ float red[2 * CH];
  const int tid  = threadIdx.x;
  const int lane = tid & 31;
  const int wv   = tid >> 5;
  const int col  = wv * 16 + (lane & 15);
  const int kr   = (lane < 16) ? 0 : 2;    // which K rows this half-wave feeds

  v2f a; a[0] = 1.0f; a[1] = 1.0f;         // ones A-matrix (16x4)
  v8f acc = {};
  for (int b = 0; b < nblk; b += 4) {
    v2f bm;
    bm[0] = partials[(b + kr)     * (2 * CH) + col];
    bm[1] = partials[(b + kr + 1) * (2 * CH) + col];
    acc = __builtin_amdgcn_wmma_f32_16x16x4_f32(
        /*neg_a=*/false, a, /*neg_b=*/false, bm,
        /*c_mod=*/(short)0, acc, /*reuse_a=*/false, /*reuse_b=*/false);
  }
  // every D row equals the column sum; row M=0 lives in acc[0], lanes 0-15
  if (lane < 16) red[col] = acc[0];
  __syncthreads();

  if (tid < CH) {
    const float inv_n = 1.0f / (float)n;
    const float mean  = red[tid] * inv_n;
    const float var   = red[CH + tid] * inv_n - mean * mean;
    const float sc    = gamma[tid] * rsqrtf(var + EPSV);
    sb[tid]      = sc;
    sb[CH + tid] = beta[tid] - mean * sc;
  }
}

// ---------------------------------------------------------------------------
// Kernel 3: streaming apply  out = out*scale[c] + bias[c]  (B128 NT traffic).
// ---------------------------------------------------------------------------
__global__ __launch_bounds__(256) void apply_bn_kernel(
    float* __restrict__ out, const float* __restrict__ sb, long total)
{
  __shared__ float ls[2 * CH];
  if (threadIdx.x < 2 * CH) ls[threadIdx.x] = sb[threadIdx.x];
  __syncthreads();

  long i = ((long)blockIdx.x * blockDim.x + threadIdx.x) * 4;
  const long stride = (long)gridDim.x * blockDim.x * 4;
  for (; i < total; i += stride) {
    v4f v = __builtin_nontemporal_load((const v4f*)&out[i]);
    const int c = (int)(i & (CH - 1));     // 4 | 64 so c..c+3 stay in-row
    v[0] = fmaf(v[0], ls[c],     ls[CH + c]);
    v[1] = fmaf(v[1], ls[c + 1], ls[CH + c + 1]);
    v[2] = fmaf(v[2], ls[c + 2], ls[CH + c + 2]);
    v[3] = fmaf(v[3], ls[c + 3], ls[CH + c + 3]);
    __builtin_nontemporal_store(v, (v4f*)&out[i]);
  }
}

extern "C" void kernel_launch(void* const* d_in, const int* in_sizes, int n_in,
                              void* d_out, int out_size, void* d_ws, size_t ws_size,
                              hipStream_t stream)
{
  const float* data  = (const float*)d_in[0];   // (N, 64)
  const float* w     = (const float*)d_in[1];   // (27, 64)
  const float* gamma = (const float*)d_in[2];   // (64,)
  const float* beta  = (const float*)d_in[3];   // (64,)
  const int*   neigh = (const int*)d_in[4];     // (N, 27)
  float* out = (float*)d_out;

  const int n = in_sizes[0] / CH;

  // workspace: partials[nblk][128] + scale/bias[128]
  int nblk = NBLK1_MAX;
  while (nblk > 4 &&
         ((size_t)nblk * (2 * CH) + 2 * CH) * sizeof(float) > ws_size)
    nblk >>= 1;
  float* partials = (float*)d_ws;
  float* sb       = partials + (size_t)nblk * (2 * CH);

  dwconv_partial_kernel<<<nblk, 256, 0, stream>>>(data, w, neigh, out,
                                                  partials, n, nblk);
  stats_wmma_kernel<<<1, 256, 0, stream>>>(partials, gamma, beta, sb, nblk, n);
  apply_bn_kernel<<<8192, 256, 0, stream>>>(out, sb, (long)n * CH);
}